// HypAgg_541165879722
// MI455X (gfx1250) — compile-verified
//
#include <hip/hip_runtime.h>
#include <cstddef>

// ---------------------------------------------------------------------------
// HypAgg for MI455X (gfx1250, wave32).
//   logmap0 -> gather/scatter spmm -> proj(expmap0)
// Memory-bound sparse workload: gather (307 MB) and 76.8M f32 atomics both hit
// the 192 MB L2 (x_tangent and accumulator are 25.6 MB each). Row norms are
// computed on the matrix pipe with V_WMMA_F32_16X16X4_F32 (square in VALU,
// reduce with a ones B-matrix), exact f32 accumulation.
// ---------------------------------------------------------------------------

typedef __attribute__((ext_vector_type(2))) float v2f;
typedef __attribute__((ext_vector_type(8))) float v8f;

#define D_FEAT 128

// --------------------------------------------------------------------------
// Zero the accumulator (graph-capture-safe replacement for hipMemsetAsync).
// --------------------------------------------------------------------------
__global__ __launch_bounds__(256) void zero_out_kernel(float4* __restrict__ p, int n4) {
    int i = blockIdx.x * blockDim.x + threadIdx.x;
    if (i < n4) {
        float4 z; z.x = 0.f; z.y = 0.f; z.z = 0.f; z.w = 0.f;
        p[i] = z;
    }
}

// --------------------------------------------------------------------------
// Per-row norm + scale. MODE 0: logmap0  (s = artanh(min(z,1-1e-7))/max(z,1e-15))
//                       MODE 1: proj∘expmap0 (s = min(tanh z, 0.996)/max(z,1e-15))
// Block = 256 threads = 8 waves; each wave owns a 16-row tile (128 rows/block).
// Phase 1: wave computes 16 row-norms^2 with 32x V_WMMA_F32_16X16X4_F32:
//   A[m,k] = src[m,k]^2 (squared in VALU), B = ones  =>  D[m][n] = ||row m||^2.
//   Loads are UNconditional (row index clamped in-bounds; norms of padding
//   rows are garbage but never read), preloaded into a register buffer so all
//   32 global_load_b64 issue before the WMMA chain (no per-load exec/waits).
//   ISA C/D layout: lane 0 VGPR j = D[j][0] (rows 0..7), lane 16 VGPR j =
//   D[8+j][0] (rows 8..15)  -> staged through LDS.
// Phase 2: coalesced float4 scale (32 lanes x 16B = one 512B row per wave).
// --------------------------------------------------------------------------
template<int MODE>
__global__ __launch_bounds__(256) void row_transform_kernel(const float* __restrict__ src,
                                                            float* __restrict__ dst,
                                                            int n_rows) {
    __shared__ float s_norm2[128];

    const int t      = threadIdx.x;
    const int lane   = t & 31;
    const int wv     = t >> 5;                 // wave in block: 0..7
    const int brow0  = blockIdx.x * 128;       // first row of this block
    const int trow0  = brow0 + wv * 16;        // first row of this wave's tile

    if (trow0 < n_rows) {                      // wave-uniform guard: EXEC all-1s for WMMA
        int m = trow0 + (lane & 15);           // A-matrix: M = lane%16
        if (m > n_rows - 1) m = n_rows - 1;    // clamp: keep address in-bounds,
                                               // padding-row norms are never read
        const int kb = (lane >> 4) << 1;       // K base: lanes 0-15 -> 0, 16-31 -> 2
        const float* rowp = src + (size_t)m * D_FEAT + kb;

        // Preload the whole row slice (32 x float2 = 64 VGPRs) so the loads
        // pipeline freely, then run the WMMA accumulate chain.
        v2f buf[32];
        #pragma unroll
        for (int i = 0; i < 32; ++i) {
            buf[i] = *(const v2f*)(rowp + i * 4);
        }

        v8f acc = {};
        v2f ones; ones.x = 1.0f; ones.y = 1.0f;
        #pragma unroll
        for (int i = 0; i < 32; ++i) {
            v2f a = buf[i];
            a.x *= a.x;
            a.y *= a.y;
            // D = A(16x4,f32) x ones(4x16) + C : D[m][*] = partial row norm^2
            acc = __builtin_amdgcn_wmma_f32_16x16x4_f32(
                false, a, false, ones, (short)0, acc, false, false);
        }

        // Stage norms^2 to LDS per the 16x16 f32 C/D layout.
        if (lane == 0) {
            #pragma unroll
            for (int j = 0; j < 8; ++j) s_norm2[wv * 16 + j] = acc[j];
        }
        if (lane == 16) {
            #pragma unroll
            for (int j = 0; j < 8; ++j) s_norm2[wv * 16 + 8 + j] = acc[j];
        }
    }
    __syncthreads();

    // Phase 2: each iteration covers 8 rows (one per wave), float4-vectorized.
    #pragma unroll 1
    for (int i = 0; i < 16; ++i) {
        const int rl = i * 8 + wv;             // local row 0..127
        const int r  = brow0 + rl;
        if (r >= n_rows) break;                // wave-uniform, monotone in i

        float z = sqrtf(s_norm2[rl]);
        z = fmaxf(z, 1e-15f);                  // MIN_NORM clamp
        float s;
        if (MODE == 0) {
            // artanh with torch-style clamp (c = 1)
            float zc = fminf(z, 1.0f - 1e-7f);
            float at = 0.5f * (log1pf(zc) - log1pf(-zc));
            s = at / z;
        } else {
            // expmap0 scale = tanh(z)/z; proj clamps ||y|| = tanh(z) to (1-4e-3)
            float th = tanhf(z);
            th = fminf(th, 0.996f);
            s = th / z;
        }

        const float4* sp = (const float4*)(src + (size_t)r * D_FEAT) + lane;
        float4 v = *sp;
        v.x *= s; v.y *= s; v.z *= s; v.w *= s;
        float4* dp = (float4*)(dst + (size_t)r * D_FEAT) + lane;
        *dp = v;
    }
}

// --------------------------------------------------------------------------
// Edge aggregation: one wave per edge. 32 lanes x float4 = one 512B row
// (coalesced global_load_b128 gather), then 4 global_atomic_add_f32 per lane
// into the L2-resident accumulator. 2-way unrolled so two independent row
// gathers are in flight per iteration; global_prefetch_b8 looks two edges
// ahead to pull rows toward the WGP.
// --------------------------------------------------------------------------
__global__ __launch_bounds__(256) void edge_agg_kernel(const float* __restrict__ xt,
                                                       const int*   __restrict__ erow,
                                                       const int*   __restrict__ ecol,
                                                       const float* __restrict__ ew,
                                                       float* __restrict__ out,
                                                       int n_edges) {
    const int lane = threadIdx.x & 31;
    const int wid  = (blockIdx.x * blockDim.x + threadIdx.x) >> 5;
    const int nw   = (gridDim.x * blockDim.x) >> 5;

    int e = wid;
    for (; e + nw < n_edges; e += 2 * nw) {
        const int e1 = e + nw;
        const int p0 = e + 2 * nw;
        const int p1 = e + 3 * nw;
        if (p0 < n_edges)
            __builtin_prefetch(xt + (size_t)ecol[p0] * D_FEAT + lane * 4, 0, 0);
        if (p1 < n_edges)
            __builtin_prefetch(xt + (size_t)ecol[p1] * D_FEAT + lane * 4, 0, 0);

        const int   r0 = erow[e];
        const int   c0 = ecol[e];
        const float w0 = ew[e];
        const int   r1 = erow[e1];
        const int   c1 = ecol[e1];
        const float w1 = ew[e1];

        float4 v0 = *((const float4*)(xt + (size_t)c0 * D_FEAT) + lane);
        float4 v1 = *((const float4*)(xt + (size_t)c1 * D_FEAT) + lane);

        float* d0 = out + (size_t)r0 * D_FEAT + (lane << 2);
        atomicAdd(d0 + 0, w0 * v0.x);
        atomicAdd(d0 + 1, w0 * v0.y);
        atomicAdd(d0 + 2, w0 * v0.z);
        atomicAdd(d0 + 3, w0 * v0.w);

        float* d1 = out + (size_t)r1 * D_FEAT + (lane << 2);
        atomicAdd(d1 + 0, w1 * v1.x);
        atomicAdd(d1 + 1, w1 * v1.y);
        atomicAdd(d1 + 2, w1 * v1.z);
        atomicAdd(d1 + 3, w1 * v1.w);
    }
    if (e < n_edges) {                          // remainder: at most one edge
        const int   r = erow[e];
        const int   c = ecol[e];
        const float w = ew[e];
        float4 v = *((const float4*)(xt + (size_t)c * D_FEAT) + lane);
        float* dp = out + (size_t)r * D_FEAT + (lane << 2);
        atomicAdd(dp + 0, w * v.x);
        atomicAdd(dp + 1, w * v.y);
        atomicAdd(dp + 2, w * v.z);
        atomicAdd(dp + 3, w * v.w);
    }
}

// --------------------------------------------------------------------------
// Launch. Inputs (setup_inputs order): x [N*128 f32], edge_row [E i32],
// edge_col [E i32], edge_weight [E f32]. Output: [N*128 f32].
// d_ws holds x_tangent (N*128*4 = 25.6 MB).
// --------------------------------------------------------------------------
extern "C" void kernel_launch(void* const* d_in, const int* in_sizes, int n_in,
                              void* d_out, int out_size, void* d_ws, size_t ws_size,
                              hipStream_t stream) {
    const float* x    = (const float*)d_in[0];
    const int*   erow = (const int*)  d_in[1];
    const int*   ecol = (const int*)  d_in[2];
    const float* ew   = (const float*)d_in[3];
    float*       out  = (float*)d_out;
    float*       xt   = (float*)d_ws;          // x_tangent scratch: n_nodes*128 f32

    const int n_nodes = in_sizes[0] / D_FEAT;
    const int n_edges = in_sizes[1];

    // 1) zero accumulator
    {
        const int n4 = n_nodes * (D_FEAT / 4);
        zero_out_kernel<<<(n4 + 255) / 256, 256, 0, stream>>>((float4*)out, n4);
    }

    // 2) logmap0: x -> xt
    const int nblk_rows = (n_nodes + 127) / 128;
    row_transform_kernel<0><<<nblk_rows, 256, 0, stream>>>(x, xt, n_nodes);

    // 3) gather * w, atomic scatter-add
    {
        int want_blocks = (n_edges + 7) / 8;   // 8 waves per block, 1 edge per wave min
        int eblocks = want_blocks < 2048 ? want_blocks : 2048;
        if (eblocks < 1) eblocks = 1;
        edge_agg_kernel<<<eblocks, 256, 0, stream>>>(xt, erow, ecol, ew, out, n_edges);
    }

    // 4) proj(expmap0(.)) in place on the accumulator
    row_transform_kernel<1><<<nblk_rows, 256, 0, stream>>>(out, out, n_nodes);
}